// GaussianWarpingScheme_36112085025359
// MI455X (gfx1250) — compile-verified
//
#include <hip/hip_runtime.h>
#include <cstdint>

// Gaussian grid sample (flow warp), B=8, C=16, H=W=256, F=3, std=0.25.
// Memory/gather-bound: WMMA not applicable (per-position value matrix => no
// shared GEMM operand). CDNA5 paths used:
//   - global_load_async_to_lds_b32 + s_wait_asynccnt (stage flow row in LDS)
//   - ds_load_b32 read-back, global_prefetch_b8 for center taps
// Key optimization: contiguous 3-wide column window + remapped weights folds
// border clamping into zero-weights, enabling b96-merged gathers (3x fewer
// vmem instructions on the hot path).

#define B_N 8
#define C_N 16
#define H_N 256
#define W_N 256
#define HW_N (H_N * W_N)

__global__ __launch_bounds__(W_N) void gauss_warp_kernel(
    const float* __restrict__ im, const float* __restrict__ w,
    float* __restrict__ out)
{
    const int wo = threadIdx.x;               // output column (0..255)
    const int ho = blockIdx.x & (H_N - 1);    // output row
    const int b  = blockIdx.x >> 8;           // batch

    __shared__ float s_flow[2 * W_N];

    // ---- Stage this block's flow row (gx, gy planes) into LDS via the
    // CDNA5 async global->LDS path. Each lane reads back only its own two
    // words, so a per-wave s_wait_asynccnt suffices (no barrier needed).
    const float* gx_src = w + (((size_t)b * 2 + 0) * H_N + ho) * W_N + wo;
    const float* gy_src = w + (((size_t)b * 2 + 1) * H_N + ho) * W_N + wo;

    // Low 32 bits of a flat pointer into LDS == wave-relative LDS byte offset.
    uint32_t lds0 = (uint32_t)(uintptr_t)&s_flow[wo];
    uint32_t lds1 = (uint32_t)(uintptr_t)&s_flow[W_N + wo];

    asm volatile(
        "global_load_async_to_lds_b32 %0, %2, off\n\t"
        "global_load_async_to_lds_b32 %1, %3, off\n\t"
        "s_wait_asynccnt 0x0"
        :
        : "v"(lds0), "v"(lds1),
          "v"((uint64_t)(uintptr_t)gx_src),
          "v"((uint64_t)(uintptr_t)gy_src)
        : "memory");

    // Plain (non-volatile) reads -> ds_load_b32; the asm memory clobber keeps
    // the compiler from folding these, and orders them after the wait.
    const float flow_x = s_flow[wo];
    const float flow_y = s_flow[W_N + wo];

    // ---- Coordinates (align_corners semantics, grid = identity - flow).
    const float bx = -1.0f + (2.0f / (float)(W_N - 1)) * (float)wo;
    const float by = -1.0f + (2.0f / (float)(H_N - 1)) * (float)ho;
    const float gx = bx - flow_x;
    const float gy = by - flow_y;

    const float x = (gx + 1.0f) * 0.5f * (float)(W_N - 1);
    const float y = (gy + 1.0f) * 0.5f * (float)(H_N - 1);

    const float xfl = floorf(x);
    const float yfl = floorf(y);
    const int   ix  = (int)xfl;
    const int   iy  = (int)yfl;
    const float fx  = x - xfl;   // in [0,1)
    const float fy  = y - yfl;

    // ---- Separable Gaussian weights: x - xi = fx - (k-1), y - yi = fy - (k-1).
    const float inv2s2 = 8.0f;   // 1 / (2 * 0.25^2)

    int   xs[3], ys[3];
    float wx[3], wy[3];
#pragma unroll
    for (int k = 0; k < 3; ++k) {
        const int xi = ix + (k - 1);
        const int yi = iy + (k - 1);
        const float dxv = fx - (float)(k - 1);
        const float dyv = fy - (float)(k - 1);
        const bool vx = (xi >= 0) && (xi < W_N);
        const bool vy = (yi >= 0) && (yi < H_N);
        xs[k] = min(max(xi, 0), W_N - 1);
        ys[k] = min(max(yi, 0), H_N - 1);
        wx[k] = vx ? expf(-dxv * dxv * inv2s2) : 0.0f;  // zero-pad via weight
        wy[k] = vy ? expf(-dyv * dyv * inv2s2) : 0.0f;
    }

    // ---- Contiguous 3-wide column window [xm, xm+2], always in-bounds.
    // Remap tap weights onto window slots: any slot collision from clamping
    // involves a weight that is exactly 0, so this is bit-identical math.
    const int xm = min(max(ix - 1, 0), W_N - 3);
    float u[3];
#pragma unroll
    for (int j = 0; j < 3; ++j) {
        float s = 0.0f;
#pragma unroll
        for (int k = 0; k < 3; ++k)
            s += ((xs[k] - xm) == j) ? wx[k] : 0.0f;
        u[j] = s;
    }

    const float* imB = im + (size_t)b * C_N * HW_N;

    // Prefetch the center-row window for a spread of channels while the exp
    // products retire (global_prefetch_b8; one line per lane cluster).
#pragma unroll
    for (int c = 0; c < C_N; c += 4)
        __builtin_prefetch(imB + c * HW_N + ys[1] * W_N + xm, 0, 3);

    // ---- Accumulate 3 rows x 3-wide window over 16 channels.
    float acc[C_N];
#pragma unroll
    for (int c = 0; c < C_N; ++c) acc[c] = 0.0f;

#pragma unroll
    for (int r = 0; r < 3; ++r) {
        const float wr = wy[r];
        if (wr != 0.0f) {    // skip fully-outside rows: saves 16 b96 gathers
            const float w0 = wr * u[0];
            const float w1 = wr * u[1];
            const float w2 = wr * u[2];
            const float* rp = imB + ys[r] * W_N + xm;
#pragma unroll
            for (int c = 0; c < C_N; ++c) {
                const float* p = rp + c * HW_N;   // 3 adjacent floats -> b96
                const float t0 = p[0];
                const float t1 = p[1];
                const float t2 = p[2];
                // dy-major / dx-minor accumulation order (matches reference;
                // zero-weight slots contribute exactly 0)
                acc[c] = fmaf(w2, t2, fmaf(w1, t1, fmaf(w0, t0, acc[c])));
            }
        }
    }

    // ---- Coalesced per-channel stores.
    float* op = out + (size_t)b * C_N * HW_N + (size_t)ho * W_N + wo;
#pragma unroll
    for (int c = 0; c < C_N; ++c) op[(size_t)c * HW_N] = acc[c];
}

extern "C" void kernel_launch(void* const* d_in, const int* in_sizes, int n_in,
                              void* d_out, int out_size, void* d_ws, size_t ws_size,
                              hipStream_t stream) {
    const float* im = (const float*)d_in[0];   // (8,16,256,256) f32
    const float* w  = (const float*)d_in[1];   // (8,2,256,256)  f32
    float* out = (float*)d_out;                // (8,16,256,256) f32
    (void)in_sizes; (void)n_in; (void)out_size; (void)d_ws; (void)ws_size;

    dim3 grid(B_N * H_N);   // 2048 blocks: one output row each
    dim3 block(W_N);        // 256 threads = 8 wave32s
    hipLaunchKernelGGL(gauss_warp_kernel, grid, block, 0, stream, im, w, out);
}